// TopKRouterCommunication_62431644615080
// MI455X (gfx1250) — compile-verified
//
#include <hip/hip_runtime.h>
#include <stdint.h>

#define B_ 8
#define N_ 4096
#define D_ 1024
#define LN_EPS 1e-5f

typedef __attribute__((ext_vector_type(2))) float v2f;
typedef __attribute__((ext_vector_type(8))) float v8f;

// ---------------------------------------------------------------------------
// Kernel 1: scores[b,n] = dot(x[b,n,:], ws) + bs       (one wave32 per row)
// ---------------------------------------------------------------------------
__global__ void k_scores(const float* __restrict__ x,
                         const float* __restrict__ wsv,
                         const float* __restrict__ bs,
                         float* __restrict__ scores) {
    const int wave = threadIdx.x >> 5;
    const int lane = threadIdx.x & 31;
    const int row  = blockIdx.x * 8 + wave;            // 0 .. B*N-1
    const float4* xr = (const float4*)(x + (size_t)row * D_);
    const float4* wr = (const float4*)wsv;
    float acc = 0.f;
#pragma unroll
    for (int w = 0; w < 8; ++w) {
        float4 a = xr[lane + 32 * w];
        float4 b = wr[lane + 32 * w];
        acc += a.x * b.x + a.y * b.y + a.z * b.z + a.w * b.w;
    }
#pragma unroll
    for (int off = 16; off > 0; off >>= 1)
        acc += __shfl_xor(acc, off, 32);
    if (lane == 0) scores[row] = acc + bs[0];
}

// ---------------------------------------------------------------------------
// Kernel 2: per-batch top-2 (value desc, index asc on ties) + xsum row.
//           xsum is 16 rows: row b = 0.5*(x[b,i0,:]+x[b,i1,:]); rows 8..15
//           are zeroed here so the WMMA kernel's A loads are branch-free.
// ---------------------------------------------------------------------------
__device__ __forceinline__ bool better(float va, int ia, float vb, int ib) {
    return (va > vb) || (va == vb && ia < ib);
}

__global__ void k_top2_xsum(const float* __restrict__ scores,
                            const unsigned char* __restrict__ alive,
                            const float* __restrict__ x,
                            int* __restrict__ topidx,
                            float* __restrict__ xsum) {
    const int b   = blockIdx.x;
    const int tid = threadIdx.x;
    __shared__ float sv0[256], sv1[256];
    __shared__ int   si0[256], si1[256];
    __shared__ int   bi0, bi1;

    const float NEG = -__builtin_huge_valf();
    float v0 = NEG, v1 = NEG;
    int   i0 = 0x7fffffff, i1 = 0x7fffffff;
    for (int n = tid; n < N_; n += 256) {
        float s = scores[(size_t)b * N_ + n];
        if (!alive[(size_t)b * N_ + n]) s = NEG;
        if (better(s, n, v0, i0)) { v1 = v0; i1 = i0; v0 = s; i0 = n; }
        else if (better(s, n, v1, i1)) { v1 = s; i1 = n; }
    }
    sv0[tid] = v0; si0[tid] = i0; sv1[tid] = v1; si1[tid] = i1;
    __syncthreads();
    for (int stride = 128; stride > 0; stride >>= 1) {
        if (tid < stride) {
            float a0 = sv0[tid], a1 = sv1[tid];
            int   ja = si0[tid], ka = si1[tid];
            float b0 = sv0[tid + stride], b1 = sv1[tid + stride];
            int   jb = si0[tid + stride], kb = si1[tid + stride];
            float r0, r1; int q0, q1;
            if (better(a0, ja, b0, jb)) {
                r0 = a0; q0 = ja;
                if (better(a1, ka, b0, jb)) { r1 = a1; q1 = ka; }
                else                        { r1 = b0; q1 = jb; }
            } else {
                r0 = b0; q0 = jb;
                if (better(a0, ja, b1, kb)) { r1 = a0; q1 = ja; }
                else                        { r1 = b1; q1 = kb; }
            }
            sv0[tid] = r0; si0[tid] = q0; sv1[tid] = r1; si1[tid] = q1;
        }
        __syncthreads();
    }
    if (tid == 0) {
        bi0 = si0[0]; bi1 = si1[0];
        topidx[b * 2 + 0] = bi0;
        topidx[b * 2 + 1] = bi1;
    }
    __syncthreads();
    const int j0 = bi0, j1 = bi1;
    const float* r0 = x + ((size_t)b * N_ + j0) * D_;
    const float* r1 = x + ((size_t)b * N_ + j1) * D_;
    for (int d = tid; d < D_; d += 256) {
        xsum[(size_t)b * D_ + d]        = 0.5f * (r0[d] + r1[d]);
        xsum[(size_t)(b + 8) * D_ + d]  = 0.0f;   // zero padding rows 8..15
    }
}

// ---------------------------------------------------------------------------
// Kernel 3: summary[b,e] = sum_d xsum[b,d] * Wr[e,d] + br[e]
//           via V_WMMA_F32_16X16X4_F32, M padded 8->16 (rows 8..15 zero in
//           memory -> unconditional A loads). One wave per 16-col tile
//           (64 waves total). Inner loop: per 32-K chunk, issue 16
//           global_load_b64 up front, then 8 chained WMMAs, so loadcnt
//           waits are staged instead of a hard wait before every WMMA.
// A layout (16x4 f32): lanes 0-15 M=lane K={0,1} in v0/v1; lanes 16-31 K={2,3}
// B layout (4x16 f32): lanes 0-15 N=lane K={0,1};          lanes 16-31 K={2,3}
// C/D layout: VGPR r -> lanes 0-15 M=r, lanes 16-31 M=r+8
// ---------------------------------------------------------------------------
__global__ void k_wmma_summary(const float* __restrict__ xsum,
                               const float* __restrict__ Wr,
                               const float* __restrict__ br,
                               float* __restrict__ summary) {
    const int wave  = threadIdx.x >> 5;
    const int lane  = threadIdx.x & 31;
    const int ntile = blockIdx.x * 8 + wave;       // 0..63, 16 output cols each
    const int m16   = lane & 15;
    const int khalf = (lane >> 4) << 1;            // 0 for lanes 0-15, 2 for 16-31

    const float* arow = xsum + (size_t)m16 * D_ + khalf;                    // 16 rows, zero-padded
    const float* brow = Wr + ((size_t)(ntile * 16 + m16)) * D_ + khalf;

    v8f c = {};
    for (int kk = 0; kk < D_; kk += 32) {
        v2f a[8], b[8];
#pragma unroll
        for (int u = 0; u < 8; ++u) {
            a[u] = *(const v2f*)(arow + kk + 4 * u);
            b[u] = *(const v2f*)(brow + kk + 4 * u);
        }
#pragma unroll
        for (int u = 0; u < 8; ++u) {
            c = __builtin_amdgcn_wmma_f32_16x16x4_f32(
                    /*neg_a=*/false, a[u], /*neg_b=*/false, b[u],
                    /*c_mod=*/(short)0, c, /*reuse_a=*/false, /*reuse_b=*/false);
        }
    }
    if (lane < 16) {
        const int ncol = ntile * 16 + lane;
        const float brv = br[ncol];
#pragma unroll
        for (int r = 0; r < 8; ++r)                 // M=r (batches 0..7)
            summary[(size_t)r * D_ + ncol] = c[r] + brv;
    }
}

// ---------------------------------------------------------------------------
// Kernel 4: out = LayerNorm(x + summary[b]) * gamma + beta  (wave32 per row,
//           h kept in registers so x is read exactly once here)
// ---------------------------------------------------------------------------
__global__ void k_layernorm(const float* __restrict__ x,
                            const float* __restrict__ summary,
                            const float* __restrict__ gamma,
                            const float* __restrict__ beta,
                            float* __restrict__ out) {
    const int wave = threadIdx.x >> 5;
    const int lane = threadIdx.x & 31;
    const int row  = blockIdx.x * 8 + wave;        // b*N + n
    const int b    = row >> 12;                    // N_ = 4096
    const float4* xr = (const float4*)(x + (size_t)row * D_);
    const float4* sr = (const float4*)(summary + (size_t)b * D_);

    float4 h[8];
    float sum = 0.f, sumsq = 0.f;
#pragma unroll
    for (int w = 0; w < 8; ++w) {
        float4 xv = xr[lane + 32 * w];
        float4 sv = sr[lane + 32 * w];
        float4 hv;
        hv.x = xv.x + sv.x; hv.y = xv.y + sv.y;
        hv.z = xv.z + sv.z; hv.w = xv.w + sv.w;
        h[w] = hv;
        sum   += hv.x + hv.y + hv.z + hv.w;
        sumsq += hv.x * hv.x + hv.y * hv.y + hv.z * hv.z + hv.w * hv.w;
    }
#pragma unroll
    for (int off = 16; off > 0; off >>= 1) {
        sum   += __shfl_xor(sum,   off, 32);
        sumsq += __shfl_xor(sumsq, off, 32);
    }
    const float inv_d = 1.0f / (float)D_;
    const float mu    = sum * inv_d;
    const float var   = sumsq * inv_d - mu * mu;
    const float rstd  = rsqrtf(var + LN_EPS);

    const float4* g4 = (const float4*)gamma;
    const float4* b4 = (const float4*)beta;
    float4* o4 = (float4*)(out + (size_t)row * D_);
#pragma unroll
    for (int w = 0; w < 8; ++w) {
        float4 g = g4[lane + 32 * w];
        float4 bb = b4[lane + 32 * w];
        float4 hv = h[w];
        float4 o;
        o.x = (hv.x - mu) * rstd * g.x + bb.x;
        o.y = (hv.y - mu) * rstd * g.y + bb.y;
        o.z = (hv.z - mu) * rstd * g.z + bb.z;
        o.w = (hv.w - mu) * rstd * g.w + bb.w;
        o4[lane + 32 * w] = o;
    }
}

// ---------------------------------------------------------------------------
extern "C" void kernel_launch(void* const* d_in, const int* in_sizes, int n_in,
                              void* d_out, int out_size, void* d_ws, size_t ws_size,
                              hipStream_t stream) {
    const float*         x     = (const float*)d_in[0];
    const unsigned char* alive = (const unsigned char*)d_in[1];
    const float*         Wr    = (const float*)d_in[2];
    const float*         br    = (const float*)d_in[3];
    const float*         wsv   = (const float*)d_in[4];
    const float*         bs    = (const float*)d_in[5];
    const float*         gamma = (const float*)d_in[6];
    const float*         beta  = (const float*)d_in[7];
    float* out = (float*)d_out;

    char*  wsb     = (char*)d_ws;
    float* scores  = (float*)wsb;                               // B*N floats
    int*   topidx  = (int*)(wsb + (size_t)B_ * N_ * 4);         // 2*B ints
    float* xsum    = (float*)(wsb + (size_t)B_ * N_ * 4 + 256); // 16*D floats (rows 8..15 zero)
    float* summary = xsum + (size_t)16 * D_;                    // B*D floats

    k_scores      <<<(B_ * N_) / 8, 256, 0, stream>>>(x, wsv, bs, scores);
    k_top2_xsum   <<<B_,            256, 0, stream>>>(scores, alive, x, topidx, xsum);
    k_wmma_summary<<<8,             256, 0, stream>>>(xsum, Wr, br, summary);
    k_layernorm   <<<(B_ * N_) / 8, 256, 0, stream>>>(x, summary, gamma, beta, out);
}